// Hyperelasticity_1219770712272
// MI455X (gfx1250) — compile-verified
//
#include <hip/hip_runtime.h>
#include <cstdint>
#include <cstddef>

#define TILE 256
#define MU_ 1.0f
#define LAM_ 1.0f

__device__ __forceinline__ void async_ld_b128(uint32_t laddr, uint32_t goff,
                                              const float* gbase) {
  asm volatile("global_load_async_to_lds_b128 %0, %1, %2"
               :: "v"(laddr), "v"(goff), "s"(gbase) : "memory");
}
__device__ __forceinline__ void async_ld_b32(uint32_t laddr, uint32_t goff,
                                             const float* gbase) {
  asm volatile("global_load_async_to_lds_b32 %0, %1, %2"
               :: "v"(laddr), "v"(goff), "s"(gbase) : "memory");
}
__device__ __forceinline__ void async_st_b128(uint32_t goff, uint32_t laddr,
                                              const float* gbase) {
  asm volatile("global_store_async_from_lds_b128 %0, %1, %2"
               :: "v"(goff), "v"(laddr), "s"(gbase) : "memory");
}
__device__ __forceinline__ void async_st_b32(uint32_t goff, uint32_t laddr,
                                             const float* gbase) {
  asm volatile("global_store_async_from_lds_b32 %0, %1, %2"
               :: "v"(goff), "v"(laddr), "s"(gbase) : "memory");
}
__device__ __forceinline__ void wait_async0() {
  asm volatile("s_wait_asynccnt 0" ::: "memory");
}

// Closed-form Neo-Hookean point update: reads 9 floats from sF[b..b+9),
// writes 9 floats each into w0/w1/w2 (PK1, PK2, sigma).
__device__ __forceinline__ void neo_point(const float* __restrict__ sF, int b,
                                          float* __restrict__ w0,
                                          float* __restrict__ w1,
                                          float* __restrict__ w2) {
  const float f00 = sF[b + 0], f01 = sF[b + 1], f02 = sF[b + 2];
  const float f10 = sF[b + 3], f11 = sF[b + 4], f12 = sF[b + 5];
  const float f20 = sF[b + 6], f21 = sF[b + 7], f22 = sF[b + 8];

  const float c00 = f11 * f22 - f12 * f21;
  const float c01 = f12 * f20 - f10 * f22;
  const float c02 = f10 * f21 - f11 * f20;
  const float det = f00 * c00 + f01 * c01 + f02 * c02;
  const float rdet = 1.0f / det;

  const float i00 = c00 * rdet;
  const float i01 = (f02 * f21 - f01 * f22) * rdet;
  const float i02 = (f01 * f12 - f02 * f11) * rdet;
  const float i10 = c01 * rdet;
  const float i11 = (f00 * f22 - f02 * f20) * rdet;
  const float i12 = (f02 * f10 - f00 * f12) * rdet;
  const float i20 = c02 * rdet;
  const float i21 = (f01 * f20 - f00 * f21) * rdet;
  const float i22 = (f00 * f11 - f01 * f10) * rdet;

  const float lnJ  = __logf(det);
  const float coef = LAM_ * lnJ - MU_;           // PK1 = mu*F + coef*F^{-T}

  const float p00 = MU_ * f00 + coef * i00;
  const float p01 = MU_ * f01 + coef * i10;
  const float p02 = MU_ * f02 + coef * i20;
  const float p10 = MU_ * f10 + coef * i01;
  const float p11 = MU_ * f11 + coef * i11;
  const float p12 = MU_ * f12 + coef * i21;
  const float p20 = MU_ * f20 + coef * i02;
  const float p21 = MU_ * f21 + coef * i12;
  const float p22 = MU_ * f22 + coef * i22;

  // M = Finv * PK1 ; PK2 = 0.5 (M + M^T)
  const float m00 = i00 * p00 + i01 * p10 + i02 * p20;
  const float m01 = i00 * p01 + i01 * p11 + i02 * p21;
  const float m02 = i00 * p02 + i01 * p12 + i02 * p22;
  const float m10 = i10 * p00 + i11 * p10 + i12 * p20;
  const float m11 = i10 * p01 + i11 * p11 + i12 * p21;
  const float m12 = i10 * p02 + i11 * p12 + i12 * p22;
  const float m20 = i20 * p00 + i21 * p10 + i22 * p20;
  const float m21 = i20 * p01 + i21 * p11 + i22 * p21;
  const float m22 = i20 * p02 + i21 * p12 + i22 * p22;
  const float q01 = 0.5f * (m01 + m10);
  const float q02 = 0.5f * (m02 + m20);
  const float q12 = 0.5f * (m12 + m21);

  // S = PK1 * F^T ; sigma = 0.5 (S + S^T) / J
  const float s00 = p00 * f00 + p01 * f01 + p02 * f02;
  const float s01 = p00 * f10 + p01 * f11 + p02 * f12;
  const float s02 = p00 * f20 + p01 * f21 + p02 * f22;
  const float s10 = p10 * f00 + p11 * f01 + p12 * f02;
  const float s11 = p10 * f10 + p11 * f11 + p12 * f12;
  const float s12 = p10 * f20 + p11 * f21 + p12 * f22;
  const float s20 = p20 * f00 + p21 * f01 + p22 * f02;
  const float s21 = p20 * f10 + p21 * f11 + p22 * f12;
  const float s22 = p20 * f20 + p21 * f21 + p22 * f22;
  const float h   = 0.5f * rdet;
  const float g01 = h * (s01 + s10);
  const float g02 = h * (s02 + s20);
  const float g12 = h * (s12 + s21);

  w0[0] = p00; w0[1] = p01; w0[2] = p02;
  w0[3] = p10; w0[4] = p11; w0[5] = p12;
  w0[6] = p20; w0[7] = p21; w0[8] = p22;
  w1[0] = m00; w1[1] = q01; w1[2] = q02;
  w1[3] = q01; w1[4] = m11; w1[5] = q12;
  w1[6] = q02; w1[7] = q12; w1[8] = m22;
  w2[0] = s00 * rdet; w2[1] = g01;        w2[2] = g02;
  w2[3] = g01;        w2[4] = s11 * rdet; w2[5] = g12;
  w2[6] = g02;        w2[7] = g12;        w2[8] = s22 * rdet;
}

// -------- hot kernel: FULL 256-point tiles only, zero data-dependent control --------
__global__ __launch_bounds__(TILE) void hyper_neo_full(
    const float* __restrict__ F, float* __restrict__ out, size_t N9) {
  __shared__ alignas(16) float sF[TILE * 9];        //  9216 B
  __shared__ alignas(16) float sOut[3][TILE * 9];   // 27648 B

  const int tid = threadIdx.x;
  const size_t tileOff = (size_t)blockIdx.x * (TILE * 9);

  // async global -> LDS: 576 b128 per tile = 2 rounds + quarter-round
  {
    const float* gbase = F + tileOff;                       // uniform
    const uint32_t lds0 = (uint32_t)(uintptr_t)(&sF[0]);
    const uint32_t o = (uint32_t)(tid << 4);
    async_ld_b128(lds0 + o,        o,        gbase);
    async_ld_b128(lds0 + o + 4096, o + 4096, gbase);
    if (tid < 64) async_ld_b128(lds0 + o + 8192, o + 8192, gbase);
    wait_async0();
  }
  __syncthreads();

  // stride-9 / stride-9 LDS access: coprime with 64 banks, conflict-free
  neo_point(sF, tid * 9, &sOut[0][tid * 9], &sOut[1][tid * 9], &sOut[2][tid * 9]);
  __syncthreads();

  // async LDS -> global: three contiguous 9216 B slices
  {
    const uint32_t o = (uint32_t)(tid << 4);
#pragma unroll
    for (int j = 0; j < 3; ++j) {
      float* obase = out + (size_t)j * N9 + tileOff;        // uniform
      const uint32_t lds0 = (uint32_t)(uintptr_t)(&sOut[j][0]);
      async_st_b128(o,        lds0 + o,        obase);
      async_st_b128(o + 4096, lds0 + o + 4096, obase);
      if (tid < 64) async_st_b128(o + 8192, lds0 + o + 8192, obase);
    }
    wait_async0();
  }
}

// -------- tail kernel: one block for the last partial tile --------
__global__ __launch_bounds__(TILE) void hyper_neo_tail(
    const float* __restrict__ F, float* __restrict__ out, int N, int tile0) {
  __shared__ alignas(16) float sF[TILE * 9];
  __shared__ alignas(16) float sOut[3][TILE * 9];

  const int tid  = threadIdx.x;
  const int npts = N - tile0;                 // 1..TILE-1
  const int nflt = npts * 9, nvec = nflt >> 2, tail = nvec << 2;

  {
    const float* gbase = F + (size_t)tile0 * 9;
    const uint32_t lds0 = (uint32_t)(uintptr_t)(&sF[0]);
    for (int k = tid; k < nvec; k += TILE) {
      uint32_t off = (uint32_t)(k << 4);
      async_ld_b128(lds0 + off, off, gbase);
    }
    for (int k = tail + tid; k < nflt; k += TILE) {
      uint32_t off = (uint32_t)(k << 2);
      async_ld_b32(lds0 + off, off, gbase);
    }
    wait_async0();
  }
  __syncthreads();

  if (tid < npts)
    neo_point(sF, tid * 9, &sOut[0][tid * 9], &sOut[1][tid * 9], &sOut[2][tid * 9]);
  __syncthreads();

  {
    const size_t N9 = (size_t)N * 9;
#pragma unroll
    for (int j = 0; j < 3; ++j) {
      float* obase = out + (size_t)j * N9 + (size_t)tile0 * 9;
      const uint32_t lds0 = (uint32_t)(uintptr_t)(&sOut[j][0]);
      for (int k = tid; k < nvec; k += TILE) {
        uint32_t off = (uint32_t)(k << 4);
        async_st_b128(off, lds0 + off, obase);
      }
      for (int k = tail + tid; k < nflt; k += TILE) {
        uint32_t off = (uint32_t)(k << 2);
        async_st_b32(off, lds0 + off, obase);
      }
    }
    wait_async0();
  }
}

extern "C" void kernel_launch(void* const* d_in, const int* in_sizes, int n_in,
                              void* d_out, int out_size, void* d_ws, size_t ws_size,
                              hipStream_t stream) {
  (void)n_in; (void)out_size; (void)d_ws; (void)ws_size;
  const float* F = (const float*)d_in[0];
  float* out = (float*)d_out;
  const int N = in_sizes[0] / 9;
  const int nfull = N / TILE;                   // full 256-point tiles
  const int rem   = N - nfull * TILE;
  if (nfull > 0)
    hyper_neo_full<<<nfull, TILE, 0, stream>>>(F, out, (size_t)N * 9);
  if (rem > 0)
    hyper_neo_tail<<<1, TILE, 0, stream>>>(F, out, N, nfull * TILE);
}